// StateDecoder_26448408609453
// MI455X (gfx1250) — compile-verified
//
#include <hip/hip_runtime.h>

// Reference: out[b, r, c] = ((x[b,c] >> r) & 1) as float32
//   B = 2048, C = 2048 (candidates), R = 32 (replicas)
// Streaming bit-expand: 16 MiB in, 512 MiB out -> store-bandwidth bound.
// MI455X floor: ~528 MiB / 23.3 TB/s ~= 23 us.
//
// Grid maps exactly: 4096 blocks x 256 threads, each block = half a row of x.
//   b    = blockIdx.x >> 1            (SALU, uniform)
//   c    = (blockIdx.x & 1)*1024 + threadIdx.x*4
// Loads/stores use SGPR base + VGPR byte offset (GVS mode) + 24-bit immediate
// offsets for the 32 per-replica stores (r*8192 B < 2^23).

#define NUM_CANDIDATES 2048
#define NUM_REPLICAS   32
#define BATCH          2048

typedef int   v4i __attribute__((ext_vector_type(4)));
typedef float v4f __attribute__((ext_vector_type(4)));

__global__ __launch_bounds__(256) void StateDecoder_bitexpand_kernel(
    const int* __restrict__ x, float* __restrict__ out)
{
    constexpr int C = NUM_CANDIDATES;   // 2048
    constexpr int R = NUM_REPLICAS;     // 32

    // Scalar (wave-uniform) part of the address: batch row + half-row select.
    const int b    = (int)(blockIdx.x >> 1);
    const int cblk = ((int)blockIdx.x & 1) * (C / 2);   // 0 or 1024

    // Lane part: 4 consecutive candidates per thread (16 B, b128-aligned).
    const int c = cblk + (int)threadIdx.x * 4;

    const int* xp = x + b * C + c;        // SGPR base (x + b*C + cblk) + VGPR lane offset

    // gfx1250 speculative prefetch of the input stream ~128 KB ahead.
    // Out-of-range speculative prefetches are silently dropped (ISA 10.5).
    __builtin_prefetch(xp + 32768, 0, 0);

    // Read-once input: non-temporal b128 load.
    const v4i xv = __builtin_nontemporal_load(reinterpret_cast<const v4i*>(xp));

    // One shared output base; the 32 stores differ only by the immediate
    // offset r*C*4 = r*8192 bytes.
    float* o = out + b * (R * C) + c;

    #pragma unroll
    for (int r = 0; r < R; ++r) {
        v4f v;
        // (x >> r) & 1 -> v_bfe_u32 ; (float)bit -> v_cvt_f32_u32 (2 VALU/elem)
        v.x = (float)(((unsigned)xv.x >> r) & 1u);
        v.y = (float)(((unsigned)xv.y >> r) & 1u);
        v.z = (float)(((unsigned)xv.z >> r) & 1u);
        v.w = (float)(((unsigned)xv.w >> r) & 1u);
        // Streaming output: non-temporal b128 store (rinse past L2).
        __builtin_nontemporal_store(v, reinterpret_cast<v4f*>(o + r * C));
    }
}

extern "C" void kernel_launch(void* const* d_in, const int* in_sizes, int n_in,
                              void* d_out, int out_size, void* d_ws, size_t ws_size,
                              hipStream_t stream)
{
    const int* x  = (const int*)d_in[0];       // [B, C] int32 bitmasks
    float*    out = (float*)d_out;             // [B, R, C] float32

    // Exact tiling: each block covers half a row (256 threads * 4 ints = 1024).
    const int total_ints = in_sizes[0];                 // B*C = 4,194,304
    const int grid       = total_ints / (NUM_CANDIDATES / 2);  // 2*B = 4096
    const int block      = 256;                         // 8 wave32 per block

    StateDecoder_bitexpand_kernel<<<grid, block, 0, stream>>>(x, out);
}